// SurfaceGraphCommunication_19628000543048
// MI455X (gfx1250) — compile-verified
//
#include <hip/hip_runtime.h>
#include <hip/hip_bf16.h>

// ---------------------------------------------------------------------------
// SurfaceGraphCommunication for MI455X (gfx1250, wave32)
//   xs = surface_x @ Ws_pre                 (WMMA f32 16x16x4 GEMM -> d_out)
//   xg = graph_x   @ Wg_pre                 (WMMA GEMM -> d_out)
//   agg_g = scatter_sum(xs[sg_src] * exp(-d^2/2s^2), sg_dst)   (fp32 atomics)
//   agg_s = scatter_sum(xg[gs_src] * exp(-d^2/2s^2), gs_dst)
//   xg_out = agg_g @ W_sg ; xs_out = agg_s @ W_gs
//   d_out.xs += xs_out @ Ws_post ; d_out.xg += xg_out @ Wg_post (in-place acc)
// ---------------------------------------------------------------------------

typedef __attribute__((ext_vector_type(2))) float v2f;
typedef __attribute__((ext_vector_type(8))) float v8f;

#define DIM 128
#define LDSP 130            // 128 + 2 floats padding -> kills LDS bank conflicts
#define SIGMA 2.5f
#define INV2S2 (1.0f / (2.0f * SIGMA * SIGMA))

// ---------------------------------------------------------------------------
// Zero-fill (float4 grid-stride)
// ---------------------------------------------------------------------------
__global__ __launch_bounds__(256) void sgc_zero_kernel(float4* __restrict__ p,
                                                       long long n4) {
  long long i = (long long)blockIdx.x * blockDim.x + threadIdx.x;
  long long stride = (long long)gridDim.x * blockDim.x;
  float4 z = make_float4(0.f, 0.f, 0.f, 0.f);
  for (; i < n4; i += stride) p[i] = z;
}

// ---------------------------------------------------------------------------
// WMMA GEMM: C[M x 128] = A[M x 128] @ W[128 x 128]   (ACCUM: C += A @ W)
// One wave -> 16 rows x 128 cols (8 accumulators of v8f, 64 VGPRs live).
// W staged transposed in LDS so each B fragment is one aligned ds_load_b64.
// Block = 256 threads = 8 waves = 128 rows.
// Ragged last tile: A-row index is CLAMPED (not predicated) -- in WMMA, A row
// m only feeds C row m, and out-of-range C rows are never stored, so garbage
// from the clamped row is harmless. This keeps the k-loop free of exec-mask
// manipulation (pure loads + WMMA).
// ---------------------------------------------------------------------------
template <bool ACCUM>
__global__ __launch_bounds__(256) void sgc_wgemm128_kernel(
    const float* __restrict__ A, const float* __restrict__ W,
    float* __restrict__ C, int M) {
  __shared__ float Wt[DIM * LDSP];  // Wt[n*LDSP + k] = W[k*128 + n]

  const int tid = threadIdx.x;
  // cooperative transposed stage of the 128x128 weight (coalesced global read;
  // LDS write stride 130 floats -> bank stride 2 -> conflict-free)
  for (int i = tid; i < DIM * DIM; i += 256) {
    int k = i >> 7;
    int n = i & 127;
    Wt[n * LDSP + k] = W[i];
  }
  __syncthreads();

  const int wave = tid >> 5;
  const int lane = tid & 31;
  const int half = lane >> 4;   // 0: K lo pair, 1: K hi pair
  const int l16  = lane & 15;
  const int row0 = (blockIdx.x * 8 + wave) * 16;
  if (row0 >= M) return;        // whole-wave early out (after barrier)

  // clamped A-fragment row (see note above) -> unconditional b64 loads
  int arow = row0 + l16;
  if (arow >= M) arow = M - 1;
  const float* __restrict__ Arow = A + (size_t)arow * DIM;

  v8f acc[8];
#pragma unroll
  for (int t = 0; t < 8; ++t) {
    if (ACCUM) {
      // preload existing C as the initial accumulator (residual / in-place);
      // clamp rows here too -- unstored rows may hold garbage.
      const int col = t * 16 + l16;
#pragma unroll
      for (int r = 0; r < 8; ++r) {
        int row = row0 + r + half * 8;
        if (row >= M) row = M - 1;
        acc[t][r] = C[(size_t)row * DIM + col];
      }
    } else {
      acc[t] = (v8f)0.0f;
    }
  }

  const float* __restrict__ Wrow = &Wt[l16 * LDSP + half * 2];
#pragma unroll 4
  for (int k = 0; k < DIM; k += 4) {
    const v2f afrag = *(const v2f*)(Arow + k + half * 2);  // aligned b64 load
#pragma unroll
    for (int t = 0; t < 8; ++t) {
      const v2f bfrag = *(const v2f*)(Wrow + t * 16 * LDSP + k);  // ds b64
      acc[t] = __builtin_amdgcn_wmma_f32_16x16x4_f32(
          /*neg_a=*/false, afrag, /*neg_b=*/false, bfrag,
          /*c_mod=*/(short)0, acc[t], /*reuse_a=*/false, /*reuse_b=*/false);
    }
  }

#pragma unroll
  for (int t = 0; t < 8; ++t) {
    const int col = t * 16 + l16;
#pragma unroll
    for (int r = 0; r < 8; ++r) {
      const int row = row0 + r + half * 8;
      if (row < M) C[(size_t)row * DIM + col] = acc[t][r];
    }
  }
}

// ---------------------------------------------------------------------------
// Edge gather/scale/scatter-sum. One wave per edge: lane l moves float4 at
// offset 4*l (coalesced 512B row), weights by exp(-d^2/(2*sigma^2)), then
// 4 hardware fp32 atomic adds into the (L2-resident) aggregation buffer.
// ---------------------------------------------------------------------------
__global__ __launch_bounds__(256) void sgc_edge_scatter_kernel(
    const float* __restrict__ feat, const float* __restrict__ dist,
    const int* __restrict__ src, const int* __restrict__ dst,
    float* __restrict__ agg, int E) {
  const int gid = blockIdx.x * blockDim.x + threadIdx.x;
  const int e = gid >> 5;
  const int lane = gid & 31;
  if (e >= E) return;

  const int s = src[e];
  const int d = dst[e];
  const float dd = dist[e];
  const float w = __expf(-(dd * dd) * INV2S2);

  const float4 v = ((const float4*)(feat + (size_t)s * DIM))[lane];
  float* out = agg + (size_t)d * DIM + lane * 4;
  unsafeAtomicAdd(out + 0, v.x * w);
  unsafeAtomicAdd(out + 1, v.y * w);
  unsafeAtomicAdd(out + 2, v.z * w);
  unsafeAtomicAdd(out + 3, v.w * w);
}

// ---------------------------------------------------------------------------
// Launcher
// ---------------------------------------------------------------------------
extern "C" void kernel_launch(void* const* d_in, const int* in_sizes, int n_in,
                              void* d_out, int out_size, void* d_ws,
                              size_t ws_size, hipStream_t stream) {
  const float* surface_x = (const float*)d_in[0];
  const float* graph_x   = (const float*)d_in[1];
  const float* sg_dist   = (const float*)d_in[2];
  const float* gs_dist   = (const float*)d_in[3];
  const float* Ws_pre    = (const float*)d_in[4];
  const float* Wg_pre    = (const float*)d_in[5];
  const float* W_sg      = (const float*)d_in[6];
  const float* W_gs      = (const float*)d_in[7];
  const float* Ws_post   = (const float*)d_in[8];
  const float* Wg_post   = (const float*)d_in[9];
  const int* sg_src      = (const int*)d_in[10];
  const int* sg_dst      = (const int*)d_in[11];
  const int* gs_src      = (const int*)d_in[12];
  const int* gs_dst      = (const int*)d_in[13];

  const int N_S  = in_sizes[0] / DIM;   // 100000
  const int N_G  = in_sizes[1] / DIM;   // 25000
  const int E_sg = in_sizes[2];         // 400000
  const int E_gs = in_sizes[3];         // 1600000

  // outputs double as xs/xg storage (residual GEMM accumulates in place)
  float* out_xs = (float*)d_out;                  // N_S*128
  float* out_xg = out_xs + (size_t)N_S * DIM;     // N_G*128

  // workspace: agg_g | agg_s | xg_out | xs_out  = 2*(N_S+N_G)*128 floats
  float* agg_g  = (float*)d_ws;
  float* agg_s  = agg_g + (size_t)N_G * DIM;
  float* xg_out = agg_s + (size_t)N_S * DIM;
  float* xs_out = xg_out + (size_t)N_G * DIM;

  const dim3 blk(256);
  const int gemm_gs = (N_S + 127) / 128;   // 128 rows per block
  const int gemm_gg = (N_G + 127) / 128;

  // 1) zero both aggregation buffers (contiguous region)
  {
    long long n4 = (long long)(N_S + N_G) * DIM / 4;
    int grid = (int)((n4 + 255) / 256);
    if (grid > 65535) grid = 65535;
    sgc_zero_kernel<<<grid, blk, 0, stream>>>((float4*)agg_g, n4);
  }

  // 2) pre-blocks: xs = surface_x@Ws_pre ; xg = graph_x@Wg_pre
  sgc_wgemm128_kernel<false><<<gemm_gs, blk, 0, stream>>>(surface_x, Ws_pre,
                                                          out_xs, N_S);
  sgc_wgemm128_kernel<false><<<gemm_gg, blk, 0, stream>>>(graph_x, Wg_pre,
                                                          out_xg, N_G);

  // 3) edge aggregation (surface->graph, graph->surface)
  sgc_edge_scatter_kernel<<<(E_sg + 7) / 8, blk, 0, stream>>>(
      out_xs, sg_dist, sg_src, sg_dst, agg_g, E_sg);
  sgc_edge_scatter_kernel<<<(E_gs + 7) / 8, blk, 0, stream>>>(
      out_xg, gs_dist, gs_src, gs_dst, agg_s, E_gs);

  // 4) bp GEMMs: xg_out = agg_g@W_sg ; xs_out = agg_s@W_gs
  sgc_wgemm128_kernel<false><<<gemm_gg, blk, 0, stream>>>(agg_g, W_sg,
                                                          xg_out, N_G);
  sgc_wgemm128_kernel<false><<<gemm_gs, blk, 0, stream>>>(agg_s, W_gs,
                                                          xs_out, N_S);

  // 5) post-blocks (residual, in place): out += X_out @ W_post
  sgc_wgemm128_kernel<true><<<gemm_gs, blk, 0, stream>>>(xs_out, Ws_post,
                                                         out_xs, N_S);
  sgc_wgemm128_kernel<true><<<gemm_gg, blk, 0, stream>>>(xg_out, Wg_post,
                                                         out_xg, N_G);
}